// triplane_hashgrid_18683107738299
// MI455X (gfx1250) — compile-verified
//
#include <hip/hip_runtime.h>

#define NPTS   1000000
#define CDIM   64
#define KDIM   192          // 3 * CDIM
#define ODIM   64
#define SZ     1024
#define HW     (SZ * SZ)

typedef float v2f __attribute__((ext_vector_type(2)));
typedef float v8f __attribute__((ext_vector_type(8)));

// ---------------------------------------------------------------------------
// Kernel 1: transpose each plane [C=64, HW] -> [HW, C=64] so that the 64
// channels of one texel are contiguous (256B).  LDS-tiled 64x64 transpose,
// coalesced on both the read and the write side.
// ---------------------------------------------------------------------------
__global__ __launch_bounds__(256)
void transpose_planes(const float* __restrict__ xy,
                      const float* __restrict__ xz,
                      const float* __restrict__ yz,
                      float* __restrict__ wsT) {
    __shared__ float t[64 * 65];           // +1 pad: conflict-free both phases
    const int tiles_per_plane = HW / 64;   // 16384
    const int pl   = blockIdx.x / tiles_per_plane;
    const int tile = blockIdx.x % tiles_per_plane;
    const float* src = (pl == 0) ? xy : ((pl == 1) ? xz : yz);
    float* dst = wsT + (size_t)pl * HW * CDIM;
    const int hw0 = tile * 64;
    const int i  = threadIdx.x & 63;       // texel within tile (read phase)
    const int cg = threadIdx.x >> 6;       // 0..3

#pragma unroll
    for (int r = 0; r < 16; ++r) {
        int c = cg * 16 + r;
        // lanes vary i -> 128B coalesced global read
        t[i * 65 + c] = src[(size_t)c * HW + hw0 + i];
    }
    __syncthreads();
#pragma unroll
    for (int r = 0; r < 16; ++r) {
        int i2 = cg * 16 + r;
        int c2 = threadIdx.x & 63;
        // lanes vary c2 -> 128B coalesced global write
        dst[(size_t)(hw0 + i2) * CDIM + c2] = t[i2 * 65 + c2];
    }
}

// ---------------------------------------------------------------------------
// Kernel 2: fused gather (bilinear sample of 3 planes) + 192->64 linear layer
// done with V_WMMA_F32_16X16X4_F32.  128 threads (4 wave32) per block,
// 64 points per block.
//   TR = true : planes are the [HW, C] transposed copies in workspace
//   TR = false: planes are the original [C, HW] inputs (fallback)
// ---------------------------------------------------------------------------
template <bool TR>
__global__ __launch_bounds__(128)
void triplane_fused(const float* __restrict__ points,   // [N,3]
                    const float* __restrict__ p0,       // xy plane
                    const float* __restrict__ p1,       // xz plane
                    const float* __restrict__ p2,       // yz plane
                    const float* __restrict__ w,        // [64,192]
                    const float* __restrict__ b,        // [64]
                    float* __restrict__ out) {          // [N,64]
    // stride 196: (196*l) mod 64 == 4*l -> all 16 A-frag lanes hit distinct
    // banks; halves offset by 2 floats cover the other 32 banks.
    __shared__ __align__(16) float feats[64 * 196];

    const int tid    = threadIdx.x;
    const int block0 = blockIdx.x * 64;

    // ---------------- gather phase ----------------
    // 64-thread group = one (point, plane) task; lane = channel.
    const int c = tid & 63;
    const int g = tid >> 6;                       // 0..1
    for (int task = g; task < 64 * 3; task += 2) {
        const int pl     = task % 3;
        const int plocal = task / 3;
        const int pt     = block0 + plocal;
        float feat = 0.f;
        if (pt < NPTS) {
            const float px = points[pt * 3 + 0];
            const float py = points[pt * 3 + 1];
            const float pz = points[pt * 3 + 2];
            const float gx = (pl == 2) ? py : px;          // xy,xz use x; yz uses y
            const float gy = (pl == 0) ? py : pz;          // xy uses y; xz,yz use z
            const float* plane = (pl == 0) ? p0 : ((pl == 1) ? p1 : p2);

            const float ix = ((gx + 1.f) * (float)SZ - 1.f) * 0.5f;
            const float iy = ((gy + 1.f) * (float)SZ - 1.f) * 0.5f;
            const float ix0f = floorf(ix), iy0f = floorf(iy);
            const float tx = ix - ix0f, ty = iy - iy0f;
            const int ix0 = (int)ix0f, iy0 = (int)iy0f;
            const int ix1 = ix0 + 1,   iy1 = iy0 + 1;
            const bool vx0 = (ix0 >= 0) & (ix0 < SZ);
            const bool vx1 = (ix1 >= 0) & (ix1 < SZ);
            const bool vy0 = (iy0 >= 0) & (iy0 < SZ);
            const bool vy1 = (iy1 >= 0) & (iy1 < SZ);

            auto ld = [&](int y, int x, bool v) -> float {
                if (!v) return 0.f;
                if (TR) return plane[((size_t)y * SZ + x) * CDIM + c]; // coalesced
                else    return plane[(size_t)c * HW + (size_t)y * SZ + x];
            };
            const float v00 = ld(iy0, ix0, vx0 & vy0);
            const float v01 = ld(iy0, ix1, vx1 & vy0);
            const float v10 = ld(iy1, ix0, vx0 & vy1);
            const float v11 = ld(iy1, ix1, vx1 & vy1);
            feat = v00 * (1.f - tx) * (1.f - ty) + v01 * tx * (1.f - ty)
                 + v10 * (1.f - tx) * ty         + v11 * tx * ty;
        }
        feats[plocal * 196 + pl * 64 + c] = feat;
    }
    __syncthreads();

    // ---------------- WMMA GEMM phase ----------------
    // wave m owns points [m*16, m*16+16); 4 accumulators cover 64 outputs.
    const int wave = tid >> 5;           // wave32
    const int lane = tid & 31;
    const int l    = lane & 15;
    const int half = lane >> 4;          // A frag: half 0 -> K,K+1 ; half 1 -> K+2,K+3

    v8f acc0 = {}, acc1 = {}, acc2 = {}, acc3 = {};
    const float* fr = &feats[(wave * 16 + l) * 196 + 2 * half];
    const float* wr = w + (size_t)l * KDIM + 2 * half;   // B frag: row n = nt*16 + l

    for (int k0 = 0; k0 < KDIM; k0 += 4) {
        v2f a  = *(const v2f*)(fr + k0);
        v2f b0 = *(const v2f*)(wr + (size_t) 0 * KDIM + k0);
        v2f b1 = *(const v2f*)(wr + (size_t)16 * KDIM + k0);
        v2f b2 = *(const v2f*)(wr + (size_t)32 * KDIM + k0);
        v2f b3 = *(const v2f*)(wr + (size_t)48 * KDIM + k0);
        acc0 = __builtin_amdgcn_wmma_f32_16x16x4_f32(false, a, false, b0, (short)0, acc0, false, false);
        acc1 = __builtin_amdgcn_wmma_f32_16x16x4_f32(false, a, false, b1, (short)0, acc1, false, false);
        acc2 = __builtin_amdgcn_wmma_f32_16x16x4_f32(false, a, false, b2, (short)0, acc2, false, false);
        acc3 = __builtin_amdgcn_wmma_f32_16x16x4_f32(false, a, false, b3, (short)0, acc3, false, false);
    }

    // C/D layout: lane = column (l), VGPR i = row i (+8 for upper half lanes)
    const int row_base = block0 + wave * 16 + half * 8;
    v8f accs[4] = {acc0, acc1, acc2, acc3};
#pragma unroll
    for (int nt = 0; nt < 4; ++nt) {
        const float bias = b[nt * 16 + l];
#pragma unroll
        for (int i = 0; i < 8; ++i) {
            const int row = row_base + i;
            if (row < NPTS)
                out[(size_t)row * ODIM + nt * 16 + l] = accs[nt][i] + bias;
        }
    }
}

// ---------------------------------------------------------------------------
extern "C" void kernel_launch(void* const* d_in, const int* in_sizes, int n_in,
                              void* d_out, int out_size, void* d_ws, size_t ws_size,
                              hipStream_t stream) {
    const float* points = (const float*)d_in[0];
    const float* xy     = (const float*)d_in[1];
    const float* xz     = (const float*)d_in[2];
    const float* yz     = (const float*)d_in[3];
    const float* w_lin  = (const float*)d_in[4];
    const float* b_lin  = (const float*)d_in[5];
    float* out = (float*)d_out;

    const size_t need = (size_t)3 * HW * CDIM * sizeof(float);  // 768 MB
    const int nblocks = (NPTS + 63) / 64;

    if (ws_size >= need) {
        float* wsT = (float*)d_ws;
        transpose_planes<<<3 * (HW / 64), 256, 0, stream>>>(xy, xz, yz, wsT);
        triplane_fused<true><<<nblocks, 128, 0, stream>>>(
            points, wsT, wsT + (size_t)HW * CDIM, wsT + (size_t)2 * HW * CDIM,
            w_lin, b_lin, out);
    } else {
        triplane_fused<false><<<nblocks, 128, 0, stream>>>(
            points, xy, xz, yz, w_lin, b_lin, out);
    }
}